// SConv3d_13142599926372
// MI455X (gfx1250) — compile-verified
//
#include <hip/hip_runtime.h>

typedef __attribute__((ext_vector_type(2))) float v2f;
typedef __attribute__((ext_vector_type(8))) float v8f;

#define GDIM 64
#define NVOX (GDIM * GDIM * GDIM)
#define INC 32
#define OUTC 64

// ---------------------------------------------------------------------------
// Kernel 1: zero the scatter workspace (sums + counts)
// ---------------------------------------------------------------------------
__global__ void zero_kernel(float* p, int n) {
    int i = blockIdx.x * blockDim.x + threadIdx.x;
    if (i < n) p[i] = 0.0f;
}

// ---------------------------------------------------------------------------
// Kernel 2: scatter-add point features into voxel sums + counts
// ---------------------------------------------------------------------------
__global__ void scatter_kernel(const float* __restrict__ pts,
                               const float* __restrict__ feats,
                               float* __restrict__ sums,
                               float* __restrict__ cnts, int npts) {
    int p = blockIdx.x * blockDim.x + threadIdx.x;
    if (p >= npts) return;
    float fx = pts[3 * p + 0];
    float fy = pts[3 * p + 1];
    float fz = pts[3 * p + 2];
    int bx = (int)floorf(fx), by = (int)floorf(fy), bz = (int)floorf(fz);
    bx = min(max(bx, 0), GDIM - 1);
    by = min(max(by, 0), GDIM - 1);
    bz = min(max(bz, 0), GDIM - 1);
    int flat = (bx * GDIM + by) * GDIM + bz;
    const float* f = feats + (long)p * INC;
    float* s = sums + (long)flat * INC;
#pragma unroll
    for (int c = 0; c < INC; ++c) atomicAdd(&s[c], f[c]);
    atomicAdd(&cnts[flat], 1.0f);
}

// ---------------------------------------------------------------------------
// Kernel 3: normalize sums -> mean (in place); empty voxels stay zero
// ---------------------------------------------------------------------------
__global__ void normalize_kernel(float* __restrict__ sums,
                                 const float* __restrict__ cnts, int n) {
    int i = blockIdx.x * blockDim.x + threadIdx.x;
    if (i >= n) return;
    float c = cnts[i / INC];
    sums[i] = sums[i] / fmaxf(c, 1.0f);
}

// ---------------------------------------------------------------------------
// Kernel 4: dense 3x3x3 conv via fp32 WMMA.
// Block = 128 threads = 4 waves; each wave owns a 16-wide OUTC tile.
// Block tile = 16 consecutive-z voxels (M). Explicit dx/dy/dz tap nest
// (no div/mod in the loop); per-tap A/B fragments are loaded into register
// arrays first (one VMEM clause), then 8 WMMAs issue back-to-back over two
// accumulators (halved D->C dependency chain).
// OOB handling: clamped address + unconditional load + v_cndmask zeroing
// (EXEC stays all-ones for WMMA).
// ---------------------------------------------------------------------------
__global__ __launch_bounds__(128) void conv_wmma_kernel(
    const float* __restrict__ voxf, const float* __restrict__ Wc,
    float* __restrict__ conv) {
    const int tile = blockIdx.x;        // 0 .. NVOX/16-1
    const int wave = threadIdx.x >> 5;  // 0..3 -> OUTC tile
    const int lane = threadIdx.x & 31;
    const int lhalf = lane >> 4;        // selects K pair within a K=4 chunk
    const int lmod = lane & 15;
    const int v0 = tile * 16;
    const int x = v0 >> 12;
    const int y = (v0 >> 6) & 63;
    const int z0 = v0 & 63;             // multiple of 16
    const int nbase = wave * 16;
    const int k0 = lhalf * 2;           // this lane's K pair offset

    v8f acc0 = {};
    v8f acc1 = {};
    const float* wt = Wc + nbase + lmod;  // advances by INC*OUTC per tap
    for (int dx = -1; dx <= 1; ++dx) {
        const int nx = x + dx;
        const bool xok = ((unsigned)nx < GDIM);
        const int cxg = min(max(nx, 0), GDIM - 1) * GDIM;
        for (int dy = -1; dy <= 1; ++dy) {
            const int ny = y + dy;
            const bool xyok = xok & ((unsigned)ny < GDIM);
            const int rowxy = (cxg + min(max(ny, 0), GDIM - 1)) * GDIM;
            for (int dz = -1; dz <= 1; ++dz) {
                const int zz = z0 + lmod + dz;  // lane's A row (voxel M=lmod)
                const bool ok = xyok & ((unsigned)zz < GDIM);
                const int cz = min(max(zz, 0), GDIM - 1);
                const float* arow = voxf + (long)(rowxy + cz) * INC + k0;

                // Load the whole tap's fragments first (single VMEM clause).
                v2f afrag[8];
                v2f bfrag[8];
#pragma unroll
                for (int kc = 0; kc < 8; ++kc) {
                    afrag[kc] = *(const v2f*)(arow + kc * 4);
                    bfrag[kc].x = wt[(long)(kc * 4 + k0) * OUTC];
                    bfrag[kc].y = wt[(long)(kc * 4 + k0 + 1) * OUTC];
                }
#pragma unroll
                for (int kc = 0; kc < 8; ++kc) {
                    v2f a = afrag[kc];
                    a.x = ok ? a.x : 0.0f;  // v_cndmask, no EXEC change
                    a.y = ok ? a.y : 0.0f;
                    if (kc & 1) {
                        acc1 = __builtin_amdgcn_wmma_f32_16x16x4_f32(
                            false, a, false, bfrag[kc], (short)0, acc1, false,
                            false);
                    } else {
                        acc0 = __builtin_amdgcn_wmma_f32_16x16x4_f32(
                            false, a, false, bfrag[kc], (short)0, acc0, false,
                            false);
                    }
                }
                wt += INC * OUTC;  // next tap's weight block
            }
        }
    }
#pragma unroll
    for (int r = 0; r < 8; ++r) {
        const int m = r + lhalf * 8;
        conv[(long)(v0 + m) * OUTC + nbase + lmod] = acc0[r] + acc1[r];
    }
}

// ---------------------------------------------------------------------------
// Kernel 5: trilinear devoxelize (8-corner gather seeds accumulator 0) fused
// with residual linear feats @ W_lin^T via fp32 WMMA (dual accumulators).
// Block = 128 threads = 4 waves, 16 points per block.
// ---------------------------------------------------------------------------
__global__ __launch_bounds__(128) void devox_wmma_kernel(
    const float* __restrict__ pts, const float* __restrict__ feats,
    const float* __restrict__ Wlin, const float* __restrict__ blin,
    const float* __restrict__ conv, const float* __restrict__ cnts,
    float* __restrict__ out, int npts) {
    __shared__ int sidx[16][8];
    __shared__ float swt[16][8];

    const int p0 = blockIdx.x * 16;
    const int tid = threadIdx.x;

    // Cooperative corner precompute: thread t -> point t/8, corner t%8.
    {
        const int pl = tid >> 3;
        const int p = min(p0 + pl, npts - 1);   // clamped, always-valid point
        const bool pvalid = (p0 + pl) < npts;
        const int c = tid & 7;
        const int dxc = (c >> 2) & 1, dyc = (c >> 1) & 1, dzc = c & 1;
        const float fx = pts[3 * p + 0];
        const float fy = pts[3 * p + 1];
        const float fz = pts[3 * p + 2];
        const int bx = (int)floorf(fx), by = (int)floorf(fy),
                  bz = (int)floorf(fz);
        const float frx = fx - (float)bx;
        const float fry = fy - (float)by;
        const float frz = fz - (float)bz;
        const int nx = bx + dxc, ny = by + dyc, nz = bz + dzc;
        const bool inb = (nx >= 0) & (nx < GDIM) & (ny >= 0) & (ny < GDIM) &
                         (nz >= 0) & (nz < GDIM);
        const int cx = min(max(nx, 0), GDIM - 1);
        const int cy = min(max(ny, 0), GDIM - 1);
        const int cz = min(max(nz, 0), GDIM - 1);
        const int fl = (cx * GDIM + cy) * GDIM + cz;
        float w = (dxc ? frx : 1.0f - frx) * (dyc ? fry : 1.0f - fry) *
                  (dzc ? frz : 1.0f - frz);
        w = (inb && pvalid) ? w : 0.0f;
        w = (cnts[fl] > 0.0f) ? w : 0.0f;
        sidx[pl][c] = fl * OUTC;
        swt[pl][c] = w;
    }
    __syncthreads();

    const int wave = tid >> 5;
    const int lane = tid & 31;
    const int lhalf = lane >> 4;
    const int lmod = lane & 15;
    const int nbase = wave * 16;
    const int n = nbase + lmod;

    // Accumulator 0 seeded with gathered trilinear contribution + bias.
    const float bias = blin[n];
    v8f acc0;
    v8f acc1 = {};
#pragma unroll
    for (int r = 0; r < 8; ++r) {
        const int m = r + lhalf * 8;
        float s = bias;
#pragma unroll
        for (int c = 0; c < 8; ++c) {
            s += swt[m][c] * conv[(long)sidx[m][c] + n];
        }
        acc0[r] = s;
    }

    // Residual GEMM: A = feats tile (16x32), B[k][n] = Wlin[n][k].
    const int pa = p0 + lmod;  // this lane's A row (point M = lmod)
    const bool pok = (pa < npts);
    const float* arow = feats + (long)min(pa, npts - 1) * INC + lhalf * 2;
    const float* brow = Wlin + (long)n * INC + lhalf * 2;
#pragma unroll
    for (int kc = 0; kc < 8; ++kc) {
        v2f a = *(const v2f*)(arow + kc * 4);  // unconditional b64 load
        a.x = pok ? a.x : 0.0f;                // v_cndmask zeroing for tail
        a.y = pok ? a.y : 0.0f;
        v2f b = *(const v2f*)(brow + kc * 4);
        if (kc & 1) {
            acc1 = __builtin_amdgcn_wmma_f32_16x16x4_f32(
                false, a, false, b, (short)0, acc1, false, false);
        } else {
            acc0 = __builtin_amdgcn_wmma_f32_16x16x4_f32(
                false, a, false, b, (short)0, acc0, false, false);
        }
    }

#pragma unroll
    for (int r = 0; r < 8; ++r) {
        const int m = r + lhalf * 8;
        if (p0 + m < npts) out[(long)(p0 + m) * OUTC + n] = acc0[r] + acc1[r];
    }
}

// ---------------------------------------------------------------------------
extern "C" void kernel_launch(void* const* d_in, const int* in_sizes, int n_in,
                              void* d_out, int out_size, void* d_ws,
                              size_t ws_size, hipStream_t stream) {
    const float* pts = (const float*)d_in[0];
    const float* feats = (const float*)d_in[1];
    const float* Wc = (const float*)d_in[2];    // (3,3,3,INC,OUTC)
    const float* Wlin = (const float*)d_in[3];  // (OUTC, INC)
    const float* blin = (const float*)d_in[4];  // (OUTC,)
    float* out = (float*)d_out;
    const int npts = in_sizes[0] / 3;

    char* ws = (char*)d_ws;
    float* sums = (float*)ws;                                   // NVOX*INC
    float* cnts = (float*)(ws + (size_t)NVOX * INC * 4);        // NVOX
    float* conv = (float*)(ws + (size_t)NVOX * (INC + 1) * 4);  // NVOX*OUTC

    // 1) zero sums + counts
    {
        const int n = NVOX * (INC + 1);
        zero_kernel<<<(n + 255) / 256, 256, 0, stream>>>(sums, n);
    }
    // 2) scatter-add
    scatter_kernel<<<(npts + 255) / 256, 256, 0, stream>>>(pts, feats, sums,
                                                           cnts, npts);
    // 3) normalize to mean
    {
        const int n = NVOX * INC;
        normalize_kernel<<<(n + 255) / 256, 256, 0, stream>>>(sums, cnts, n);
    }
    // 4) dense 3x3x3 conv via WMMA (16 voxels per block, 4 OUTC tiles)
    conv_wmma_kernel<<<NVOX / 16, 128, 0, stream>>>(sums, Wc, conv);
    // 5) devoxelize + residual linear via WMMA
    devox_wmma_kernel<<<(npts + 15) / 16, 128, 0, stream>>>(
        pts, feats, Wlin, blin, conv, cnts, out, npts);
}